// NormalsRenderer_70755291234864
// MI455X (gfx1250) — compile-verified
//
#include <hip/hip_runtime.h>
#include <hip/hip_bf16.h>
#include <math.h>

typedef __attribute__((ext_vector_type(2))) float v2f;
typedef __attribute__((ext_vector_type(8))) float v8f;

#define BATCH 512
#define NPTS  512

__device__ __forceinline__ float pair_w(float c)
{
    // exp(-arccos(clip(c, -1, 1))) — branch-free
    float cv = fminf(fmaxf(c, -1.0f), 1.0f);
    return __expf(-acosf(cv));
}

// ---------------------------------------------------------------------------
// Kernel 1: per (batch, 16-row block) compute row sums of
//   w_pair[i][j] = (i==j) ? 0 : exp(-arccos(clip(dot(n_i, n_j), -1, 1)))
// via V_WMMA_F32_16X16X4_F32 (K=3 padded to 4) for the cosine tiles.
// The inner loop is fully branch-free (no diagonal masking): we sum over ALL j
// and subtract the analytically known diagonal term w(i,i) once at the end.
// One wave32 per 16-row block; 8 waves per workgroup; 4 workgroups per batch.
// ---------------------------------------------------------------------------
__global__ __launch_bounds__(256) void k_rowsum(const float* __restrict__ normals,
                                                float* __restrict__ S)
{
    __shared__ float lds[NPTS * 4];          // xyz0-padded normals (8 KB)

    const int b    = blockIdx.x >> 2;        // 4 blocks per batch
    const int ig   = blockIdx.x & 3;
    const int tid  = threadIdx.x;
    const int wave = tid >> 5;
    const int lane = tid & 31;
    const int m    = lane & 15;              // row/col within 16-tile
    const int h    = lane >> 4;              // half-wave: K={0,1} vs K={2,3}

    const float* nb = normals + (size_t)b * NPTS * 3;
    for (int i = tid; i < NPTS; i += 256) {
        lds[i * 4 + 0] = nb[i * 3 + 0];
        lds[i * 4 + 1] = nb[i * 3 + 1];
        lds[i * 4 + 2] = nb[i * 3 + 2];
        lds[i * 4 + 3] = 0.0f;               // K-pad -> exact 3-component dot
    }
    __syncthreads();

    const int i_blk = ig * 8 + wave;
    const int i0    = i_blk * 16;

    // A tile (16x4): lanes 0-15 hold K=0,1 in v0,v1; lanes 16-31 hold K=2,3.
    v2f a;
    a.x = lds[(i0 + m) * 4 + 2 * h + 0];
    a.y = lds[(i0 + m) * 4 + 2 * h + 1];

    float rowAcc[8];
#pragma unroll
    for (int r = 0; r < 8; ++r) rowAcc[r] = 0.0f;

    // Branch-free hot loop: ds_load -> wmma -> 8x (clamp, acos, exp, add).
    // Unroll x2 so the next tile's 8-byte LDS load overlaps this tile's VALU.
#pragma unroll 2
    for (int jb = 0; jb < 32; ++jb) {
        const int j0 = jb * 16;
        v2f bm;
        bm.x = lds[(j0 + m) * 4 + 2 * h + 0];
        bm.y = lds[(j0 + m) * 4 + 2 * h + 1];

        v8f c = {};
        c = __builtin_amdgcn_wmma_f32_16x16x4_f32(false, a, false, bm,
                                                  (short)0, c, false, false);
#pragma unroll
        for (int r = 0; r < 8; ++r)
            rowAcc[r] += pair_w(c[r]);
    }

    // Sum each row across the 16 lanes of its half-wave.
#pragma unroll
    for (int r = 0; r < 8; ++r) {
        float v = rowAcc[r];
        v += __shfl_xor(v, 8, 32);
        v += __shfl_xor(v, 4, 32);
        v += __shfl_xor(v, 2, 32);
        v += __shfl_xor(v, 1, 32);
        rowAcc[r] = v;
    }

    // Lane m==0 of each half-wave holds rows i0 + r + 8*h. Subtract the
    // diagonal term w(i,i) = exp(-acos(clip(||n_i||^2))) that the unmasked
    // sum included, then store.
    if (m == 0) {
#pragma unroll
        for (int r = 0; r < 8; ++r) {
            const int row = i0 + r + 8 * h;
            const float x = lds[row * 4 + 0];
            const float y = lds[row * 4 + 1];
            const float z = lds[row * 4 + 2];
            const float wii = pair_w(x * x + y * y + z * z);
            S[(size_t)b * NPTS + row] = rowAcc[r] - wii;
        }
    }
}

// ---------------------------------------------------------------------------
// Kernel 2: global max over all B*N row sums (single workgroup, deterministic).
// ---------------------------------------------------------------------------
__global__ __launch_bounds__(1024) void k_max(const float* __restrict__ S,
                                              float* __restrict__ outMax)
{
    __shared__ float red[1024];
    float mx = 0.0f;                          // row sums are >= 0
    for (int i = threadIdx.x; i < BATCH * NPTS; i += 1024)
        mx = fmaxf(mx, S[i]);
    red[threadIdx.x] = mx;
    __syncthreads();
    for (int s = 512; s > 0; s >>= 1) {
        if (threadIdx.x < (unsigned)s)
            red[threadIdx.x] = fmaxf(red[threadIdx.x], red[threadIdx.x + s]);
        __syncthreads();
    }
    if (threadIdx.x == 0) outMax[0] = red[0];
}

// ---------------------------------------------------------------------------
// Kernel 3: per-batch weighted mean normal + safe normalize.
// ---------------------------------------------------------------------------
__global__ __launch_bounds__(256) void k_final(const float* __restrict__ S,
                                               const float* __restrict__ maxbuf,
                                               const float* __restrict__ normals,
                                               float* __restrict__ out)
{
    __shared__ float rx[256], ry[256], rz[256];
    const int b   = blockIdx.x;
    const int tid = threadIdx.x;
    const float inv = 1.0f / maxbuf[0];

    float sx = 0.0f, sy = 0.0f, sz = 0.0f;
    for (int i = tid; i < NPTS; i += 256) {
        const float w  = S[(size_t)b * NPTS + i] * inv;
        const float* p = normals + ((size_t)b * NPTS + i) * 3;
        sx += w * p[0];
        sy += w * p[1];
        sz += w * p[2];
    }
    rx[tid] = sx; ry[tid] = sy; rz[tid] = sz;
    __syncthreads();
    for (int s = 128; s > 0; s >>= 1) {
        if (tid < (unsigned)s) {
            rx[tid] += rx[tid + s];
            ry[tid] += ry[tid + s];
            rz[tid] += rz[tid + s];
        }
        __syncthreads();
    }
    if (tid == 0) {
        float nx = rx[0], ny = ry[0], nz = rz[0];
        float nrm2 = fmaxf(nx * nx + ny * ny + nz * nz, 1e-20f);
        float invn = 1.0f / sqrtf(nrm2);
        out[b * 3 + 0] = nx * invn;
        out[b * 3 + 1] = ny * invn;
        out[b * 3 + 2] = nz * invn;
    }
}

// ---------------------------------------------------------------------------
// Launch: d_in[0] = normals [512,512,3] f32 (d_in[1] weights is unused by the
// reference math). d_out = [512,3] f32. Workspace: S (1 MB) + 1 float max.
// ---------------------------------------------------------------------------
extern "C" void kernel_launch(void* const* d_in, const int* in_sizes, int n_in,
                              void* d_out, int out_size, void* d_ws, size_t ws_size,
                              hipStream_t stream)
{
    (void)in_sizes; (void)n_in; (void)out_size; (void)ws_size;

    const float* normals = (const float*)d_in[0];
    float* out    = (float*)d_out;
    float* S      = (float*)d_ws;                 // BATCH*NPTS floats
    float* maxbuf = S + (size_t)BATCH * NPTS;     // 1 float

    k_rowsum<<<BATCH * 4, 256, 0, stream>>>(normals, S);
    k_max  <<<1,         1024, 0, stream>>>(S, maxbuf);
    k_final<<<BATCH,      256, 0, stream>>>(S, maxbuf, normals, out);
}